// _ProposalLayer_103079215569
// MI455X (gfx1250) — compile-verified
//
#include <hip/hip_runtime.h>
#include <stdint.h>

// ---- problem constants (match reference) ----
#define BATCH 8
#define HH 512
#define WW 512
#define HWSZ (HH * WW)          // 262144, power of two
#define HBINS 16384             // bins = float_bits >> 16; scores < 1.0 => bin < 0x3F80 = 16256
#define CAP 8192                // per-batch candidate capacity (K=2000 + bin ties << CAP)
#define TILE 64
#define TDIM 66                 // TILE + 2 halo
#define FEAT_STRIDE_F 16.0f

typedef unsigned int u32;
typedef unsigned long long u64;

// ---- CDNA5 async global->LDS copy path (guarded; fallback = plain LDS store) ----
#if defined(__has_builtin)
#  if __has_builtin(__builtin_amdgcn_global_load_async_to_lds_b32)
#    define HAVE_ASYNC_LDS 1
#  endif
#endif
#ifndef HAVE_ASYNC_LDS
#  define HAVE_ASYNC_LDS 0
#endif

typedef __attribute__((address_space(3))) int lds_i32;
typedef __attribute__((address_space(1))) int glb_i32;

__device__ __forceinline__ void copy_g2lds(float* dst, const float* src) {
#if HAVE_ASYNC_LDS
  // emits global_load_async_to_lds_b32 (tracked by ASYNCcnt)
  __builtin_amdgcn_global_load_async_to_lds_b32((glb_i32*)src, (lds_i32*)dst, 0, 0);
#else
  *dst = *src;
#endif
}

__device__ __forceinline__ void wait_lds_copies() {
#if HAVE_ASYNC_LDS
#  if __has_builtin(__builtin_amdgcn_s_wait_asynccnt)
  __builtin_amdgcn_s_wait_asynccnt(0);
#  else
  asm volatile("s_wait_asynccnt 0" ::: "memory");
#  endif
#endif
}

// ---- kernel 0: zero histogram + candidate counters (must be per-call) ----
__global__ void zero_u32(u32* __restrict__ p, int n) {
  int i = blockIdx.x * blockDim.x + threadIdx.x;
  if (i < n) p[i] = 0u;
}

// ---- kernels 1 & 3: tiled 3x3 peak-NMS; MODE 0 = histogram, MODE 1 = compact ----
template <int MODE>
__global__ void nms_pass(const float* __restrict__ scores,
                         u32* __restrict__ hist,
                         const u32* __restrict__ thr,
                         u32* __restrict__ cnt,
                         u64* __restrict__ cand) {
  __shared__ float tile[TDIM * TDIM];  // ~17 KB

  const int bid = blockIdx.x;
  const int b   = bid >> 6;        // 64 tiles per batch (8x8)
  const int t   = bid & 63;
  const int ty  = (t >> 3) * TILE;
  const int tx  = (t & 7) * TILE;
  const float* sb = scores + (size_t)b * HWSZ;
  const int tid = threadIdx.x;

  // Stage TILE+halo into LDS via async DMA (OOB halo -> 0.0f; scores are >0 so
  // a 0 pad can never equal/beat an in-bounds peak — equivalent to -inf pad).
  for (int i = tid; i < TDIM * TDIM; i += blockDim.x) {
    const int ly = i / TDIM, lx = i % TDIM;
    const int gy = ty + ly - 1, gx = tx + lx - 1;
    if (gy >= 0 && gy < HH && gx >= 0 && gx < WW) {
      copy_g2lds(&tile[i], &sb[gy * WW + gx]);
    } else {
      tile[i] = 0.0f;
    }
  }
  wait_lds_copies();
  __syncthreads();

  u32 T = 0;
  if (MODE == 1) T = thr[b];

  for (int p = tid; p < TILE * TILE; p += blockDim.x) {
    const int py = p >> 6, px = p & 63;
    const float* c = &tile[(py + 1) * TDIM + (px + 1)];
    const float v = c[0];
    float m = v;
    m = fmaxf(m, c[-TDIM - 1]); m = fmaxf(m, c[-TDIM]); m = fmaxf(m, c[-TDIM + 1]);
    m = fmaxf(m, c[-1]);                                 m = fmaxf(m, c[1]);
    m = fmaxf(m, c[TDIM - 1]);  m = fmaxf(m, c[TDIM]);   m = fmaxf(m, c[TDIM + 1]);
    if (v == m) {  // local maximum survives NMS
      const u32 bits = __float_as_uint(v);
      const u32 bin  = bits >> 16;  // monotonic key for positive floats
      if (MODE == 0) {
        atomicAdd(&hist[b * HBINS + bin], 1u);
      } else if (bin >= T) {
        const u32 pos = atomicAdd(&cnt[b], 1u);
        if (pos < CAP) {
          const u32 gidx = (u32)((ty + py) * WW + (tx + px));
          // primary: score desc; secondary: index asc (matches jax top_k ties)
          cand[(size_t)b * CAP + pos] = ((u64)bits << 32) | (u64)(0xFFFFFFFFu - gidx);
        }
      }
    }
  }
}

// ---- kernel 2: per-batch radix-select threshold bin ----
__global__ void find_threshold(const u32* __restrict__ hist,
                               const int* __restrict__ Kp,
                               u32* __restrict__ thr) {
  __shared__ u32 part[256];
  const int b = blockIdx.x;
  const int tid = threadIdx.x;
  const u32* hb = hist + b * HBINS;
  const int CH = HBINS / 256;  // 64 bins per thread
  u32 s = 0;
  const int base = tid * CH;
  for (int i = 0; i < CH; ++i) s += hb[base + i];
  part[tid] = s;
  __syncthreads();
  if (tid == 0) {
    int Ki = *Kp; if (Ki < 1) Ki = 1;
    const u32 K = (u32)Ki;
    u32 cum = 0;
    int tsel = -1;
    for (int tt = 255; tt >= 0; --tt) {
      if (cum + part[tt] >= K) { tsel = tt; break; }
      cum += part[tt];
    }
    u32 T = 0;
    if (tsel >= 0) {
      const int lo = tsel * CH;
      for (int bin = lo + CH - 1; bin >= lo; --bin) {
        const u32 c = hb[bin];
        if (cum + c >= K) { T = (u32)bin; break; }
        cum += c;
      }
    }
    thr[b] = T;  // collect everything with bin >= T  (=> at least K candidates)
  }
}

// ---- kernel 4: per-batch bitonic sort (desc) in LDS + box emit ----
__global__ __launch_bounds__(1024)
void sort_emit(const u64* __restrict__ cand, const u32* __restrict__ cnt,
               const int* __restrict__ Kp,
               const float* __restrict__ wh, const float* __restrict__ off,
               float* __restrict__ out) {
  __shared__ u64 keys[CAP];  // 64 KB LDS
  const int b = blockIdx.x;
  const int tid = threadIdx.x;
  u32 m = cnt[b]; if (m > CAP) m = CAP;
  const u64* cb = cand + (size_t)b * CAP;
  for (int i = tid; i < CAP; i += 1024) keys[i] = (i < (int)m) ? cb[i] : 0ull;
  __syncthreads();

  for (u32 k = 2; k <= CAP; k <<= 1) {
    for (u32 j = k >> 1; j > 0; j >>= 1) {
      for (u32 i = (u32)tid; i < CAP; i += 1024) {
        const u32 ixj = i ^ j;
        if (ixj > i) {
          const u64 a = keys[i], c = keys[ixj];
          const bool desc = ((i & k) == 0);
          if (desc ? (a < c) : (a > c)) { keys[i] = c; keys[ixj] = a; }
        }
      }
      __syncthreads();
    }
  }

  int K = *Kp; if (K < 0) K = 0;
  const int lim = K < CAP ? K : CAP;
  const float* whb  = wh  + (size_t)b * 2 * HWSZ;
  const float* offb = off + (size_t)b * 2 * HWSZ;
  for (int kk = tid; kk < lim; kk += 1024) {
    const u64 key = keys[kk];
    u32 idx = 0xFFFFFFFFu - (u32)(key & 0xFFFFFFFFu);
    idx &= (HWSZ - 1);  // safety clamp for pad keys (shouldn't occur: ~29k peaks >> K)
    const float xs = (float)(idx & (WW - 1));
    const float ys = (float)(idx >> 9);         // / 512
    const float ox = offb[idx];
    const float oy = offb[HWSZ + idx];
    const float w0 = whb[idx];
    const float h0 = whb[HWSZ + idx];
    const float cx = xs + ox, cy = ys + oy;
    float* o = out + ((size_t)b * (size_t)K + (size_t)kk) * 5;
    o[0] = (float)b;
    o[1] = (cx - 0.5f * w0) * FEAT_STRIDE_F;
    o[2] = (cy - 0.5f * h0) * FEAT_STRIDE_F;
    o[3] = (cx + 0.5f * w0) * FEAT_STRIDE_F;
    o[4] = (cy + 0.5f * h0) * FEAT_STRIDE_F;
  }
}

extern "C" void kernel_launch(void* const* d_in, const int* in_sizes, int n_in,
                              void* d_out, int out_size, void* d_ws, size_t ws_size,
                              hipStream_t stream) {
  const float* scores = (const float*)d_in[0];
  const float* wh     = (const float*)d_in[1];
  const float* off    = (const float*)d_in[2];
  const int*   Kp     = (const int*)d_in[3];
  float* out = (float*)d_out;

  // workspace layout (u32 units): [hist: B*HBINS][cnt: B][thr: B][cand: B*CAP u64]
  u32* ws   = (u32*)d_ws;
  u32* hist = ws;
  u32* cnt  = ws + BATCH * HBINS;
  u32* thr  = ws + BATCH * HBINS + BATCH;
  u64* cand = (u64*)(ws + BATCH * HBINS + 2 * BATCH);  // byte offset 524352, 8B aligned

  const int nz = BATCH * HBINS + BATCH;  // hist + cnt are contiguous
  hipLaunchKernelGGL(zero_u32, dim3((nz + 255) / 256), dim3(256), 0, stream, hist, nz);
  hipLaunchKernelGGL((nms_pass<0>), dim3(BATCH * 64), dim3(256), 0, stream,
                     scores, hist, thr, cnt, cand);
  hipLaunchKernelGGL(find_threshold, dim3(BATCH), dim3(256), 0, stream, hist, Kp, thr);
  hipLaunchKernelGGL((nms_pass<1>), dim3(BATCH * 64), dim3(256), 0, stream,
                     scores, hist, thr, cnt, cand);
  hipLaunchKernelGGL(sort_emit, dim3(BATCH), dim3(1024), 0, stream,
                     cand, cnt, Kp, wh, off, out);
}